// HRM_5961414607316
// MI455X (gfx1250) — compile-verified
//
// HRM GNN forward for MI455X (gfx1250, wave32, WMMA).
// Dense 128-col GEMMs run on V_WMMA_F32_16X16X4_F32 (exact fp32, matches reference).
// Edge softmax-aggregation uses atomicMax (ordered-uint encoding) + atomicAdd f32.
//
// Input layout (setup_inputs dict order, params pytree flattened jax-style:
// dict keys sorted, lists in order):
//   d_in[0] x_cell [N,16] f32       d_in[1..3] ei_line/region/diag [2,E] i32
//   d_in[4] batch_size (unused on host; B=50 per module constant)
//   d_in[5+i] = params leaf i:
//    0 embed.W 1 embed.b 2 embed.scale
//    3 hmod.Wk 4 hmod.Wm1 5 hmod.Wm2 6 hmod.Wq 7 hmod.Wv 8 hmod.bk 9 hmod.bm1
//   10 hmod.bm2 11 hmod.bq 12 hmod.bv 13 hmod.scale
//   14 lblock.Wi 15 lblock.bi
//   16 film.W1 17 film.W2 18 film.b1 19 film.b2
//   20..31 gat1[r]{W,att_dst,att_src,bias} r=0..2
//   32..43 gat2[r]{W,att_dst,att_src,bias}
//   44 hgt.Wa 45 hgt.Wk 46 hgt.Wq 47 hgt.Wv 48 hgt.a_rel 49 hgt.ba 50 hgt.bk
//   51 hgt.bq 52 hgt.bv 53 hgt.m_rel 54 hgt.p_rel 55 hgt.skip
//   56 lblock.s1 57 lblock.s2 58 lblock.s_hgt
//   59 readout.W1 60 readout.W2 61 readout.b1 62 readout.b2
//   63 z0
#include <hip/hip_runtime.h>
#include <math.h>

typedef __attribute__((ext_vector_type(2))) float v2f;
typedef __attribute__((ext_vector_type(8))) float v8f;

#define HD 128            // d_model
#define GAT_H 2
#define HGT_H 4
#define EPS_RMS 1e-6f
#define INV_SQRT32 0.17677669529663687f
#define NEG_INF_ENC 0x007FFFFFu   // ordered encoding of -inf

__device__ __forceinline__ float lrelu_f(float x) { return x > 0.f ? x : 0.2f * x; }
__device__ __forceinline__ float gelu_f(float x) { return 0.5f * x * (1.f + erff(x * 0.70710678118654752f)); }
__device__ __forceinline__ unsigned enc_f(float x) {
  unsigned u = __float_as_uint(x);
  return (u & 0x80000000u) ? ~u : (u | 0x80000000u);
}
__device__ __forceinline__ float dec_f(unsigned u) {
  u = (u & 0x80000000u) ? (u ^ 0x80000000u) : ~u;
  return __uint_as_float(u);
}

// ---------------- WMMA GEMM: Y[N,128] = op(X[N,K]) @ W[128,K]^T (+bias) ----------------
// 8 waves/block; wave w -> rows [blk*128+16w, +16), all 8 col-tiles (8x v8f acc).
// Fragment layouts per CDNA5 ISA 7.12.2 (32-bit A 16x4 / B 4x16 / D 16x16).
// Out-of-range rows are CLAMPED (not masked): their D rows are never stored, so the
// K-loop is completely branch-free (no per-step exec-mask juggling).
#define MBIAS 1
#define MACC 2
#define MGELUA 4
#define MGELUO 8

template <int K, int MODE>
__global__ __launch_bounds__(256) void k_gemm(const float* __restrict__ X,
                                              const float* __restrict__ W,
                                              const float* __restrict__ bias,
                                              float* __restrict__ Y, int N) {
  const int lane = threadIdx.x & 31;
  const int wave = threadIdx.x >> 5;
  const int rowBase = blockIdx.x * 128 + wave * 16;
  const int half = lane >> 4;  // 0: K+0/1, 1: K+2/3
  const int lr = lane & 15;
  int arow = rowBase + lr;
  if (arow >= N) arow = N - 1;  // clamp: stays in-bounds, result rows unused
  const float* Xr = X + (size_t)arow * K + 2 * half;
  const float* Wr = W + (size_t)lr * K + 2 * half;

  v8f acc[8] = {};

  for (int k0 = 0; k0 < K; k0 += 4) {
    v2f a = *(const v2f*)(Xr + k0);  // 8B-aligned (k0+2*half even)
    if constexpr ((MODE & MGELUA) != 0) { a.x = gelu_f(a.x); a.y = gelu_f(a.y); }
#pragma unroll
    for (int c = 0; c < 8; ++c) {
      v2f b = *(const v2f*)(Wr + (size_t)(c * 16) * K + k0);  // B[k][n] = W[n][k]
      acc[c] = __builtin_amdgcn_wmma_f32_16x16x4_f32(false, a, false, b, (short)0,
                                                     acc[c], false, false);
    }
  }

  const bool fullTile = (rowBase + 16) <= N;
#pragma unroll
  for (int c = 0; c < 8; ++c) {
    const int col = c * 16 + lr;
    float bv = 0.f;
    if constexpr ((MODE & MBIAS) != 0) bv = bias[col];
    float* p0 = Y + (size_t)(rowBase + half * 8) * HD + col;
    if (fullTile) {
#pragma unroll
      for (int j = 0; j < 8; ++j) {
        float v = acc[c][j] + bv;
        if constexpr ((MODE & MGELUO) != 0) v = gelu_f(v);
        float* p = p0 + (size_t)j * HD;
        if constexpr ((MODE & MACC) != 0) *p += v; else *p = v;
      }
    } else {
#pragma unroll
      for (int j = 0; j < 8; ++j) {
        const int row = rowBase + half * 8 + j;
        if (row < N) {
          float v = acc[c][j] + bv;
          if constexpr ((MODE & MGELUO) != 0) v = gelu_f(v);
          float* p = p0 + (size_t)j * HD;
          if constexpr ((MODE & MACC) != 0) *p += v; else *p = v;
        }
      }
    }
  }
}

// Block-diagonal GEMM for HGT rel-einsum: out[n,h*32+e] = sum_d in[n,h*32+d]*A[h,d,e]
// A = a_rel[r] or m_rel[r], laid out [H][32][32]. 8 waves/block, 16 rows/block:
// wave w -> head w/2, col-tile w%2, K=32 (8 wmma steps). Branch-free K-loop and
// fast-path epilogue (all tiles full when N % 16 == 0).
__global__ __launch_bounds__(256) void k_gemm_bd(const float* __restrict__ X,
                                                 const float* __restrict__ A,
                                                 float* __restrict__ Y, int N) {
  const int lane = threadIdx.x & 31;
  const int wave = threadIdx.x >> 5;
  const int head = wave >> 1;
  const int ct = wave & 1;
  const int rowBase = blockIdx.x * 16;
  const int half = lane >> 4;
  const int lr = lane & 15;
  int arow = rowBase + lr;
  if (arow >= N) arow = N - 1;
  const float* Xr = X + (size_t)arow * HD + head * 32 + 2 * half;
  const float* Ah = A + head * 32 * 32 + 2 * half * 32 + ct * 16 + lr;

  v8f acc = {};
#pragma unroll
  for (int k0 = 0; k0 < 32; k0 += 4) {
    v2f a = *(const v2f*)(Xr + k0);
    const float* Bp = Ah + k0 * 32;  // B[d][e] = A[h][d][e]
    v2f b; b.x = Bp[0]; b.y = Bp[32];
    acc = __builtin_amdgcn_wmma_f32_16x16x4_f32(false, a, false, b, (short)0, acc,
                                                false, false);
  }
  const int col = head * 32 + ct * 16 + lr;
  float* p0 = Y + (size_t)(rowBase + half * 8) * HD + col;
  if (rowBase + 16 <= N) {
#pragma unroll
    for (int j = 0; j < 8; ++j) p0[(size_t)j * HD] = acc[j];
  } else {
#pragma unroll
    for (int j = 0; j < 8; ++j) {
      const int row = rowBase + half * 8 + j;
      if (row < N) p0[(size_t)j * HD] = acc[j];
    }
  }
}

// ---------------- elementwise / fill ----------------
__global__ void k_fill_u32(unsigned* p, unsigned v, int n) {
  int i = blockIdx.x * blockDim.x + threadIdx.x;
  if (i < n) p[i] = v;
}
__global__ void k_fill_f32(float* p, float v, size_t n) {
  size_t i = (size_t)blockIdx.x * blockDim.x + threadIdx.x;
  if (i < n) p[i] = v;
}
__global__ void k_bias3(const float* b0, const float* b1, const float* b2,
                        float* acc, size_t tot) {
  size_t i = (size_t)blockIdx.x * blockDim.x + threadIdx.x;
  if (i < tot) { int c = (int)(i & (HD - 1)); acc[i] = b0[c] + b1[c] + b2[c]; }
}
__global__ void k_zinit(float* z, const float* z0, int n) {
  int i = blockIdx.x * blockDim.x + threadIdx.x;
  if (i < n) z[i] = z0[i & (HD - 1)];
}

// ---------------- embed: xemb = rms(gelu(xc @ We^T + be), scale) ----------------
__global__ __launch_bounds__(128) void k_embed(const float* __restrict__ xc,
                                               const float* __restrict__ We,
                                               const float* __restrict__ be,
                                               const float* __restrict__ scale,
                                               float* __restrict__ xemb, int N) {
  const int n = blockIdx.x, o = threadIdx.x;
  __shared__ float xs[16];
  __shared__ float red[128];
  if (o < 16) xs[o] = xc[(size_t)n * 16 + o];
  __syncthreads();
  float s = be[o];
#pragma unroll
  for (int i = 0; i < 16; ++i) s += xs[i] * We[o * 16 + i];
  float g = gelu_f(s);
  red[o] = g * g;
  __syncthreads();
  for (int st = 64; st > 0; st >>= 1) { if (o < st) red[o] += red[o + st]; __syncthreads(); }
  float r = rsqrtf(red[0] / 128.f + EPS_RMS);
  xemb[(size_t)n * HD + o] = scale[o] * g * r;
}

// x_out = lrelu(x_in + scale * acc * rsqrt(mean(acc^2)+eps))
__global__ __launch_bounds__(128) void k_rms_res_lrelu(const float* __restrict__ xin,
                                                       const float* __restrict__ acc,
                                                       const float* __restrict__ scale,
                                                       float* __restrict__ xout, int N) {
  const int n = blockIdx.x, o = threadIdx.x;
  __shared__ float red[128];
  float a = acc[(size_t)n * HD + o];
  red[o] = a * a;
  __syncthreads();
  for (int st = 64; st > 0; st >>= 1) { if (o < st) red[o] += red[o + st]; __syncthreads(); }
  float r = rsqrtf(red[0] / 128.f + EPS_RMS);
  xout[(size_t)n * HD + o] = lrelu_f(xin[(size_t)n * HD + o] + scale[o] * a * r);
}

// HGT tail: t = g*y + (1-g)*x ; x = lrelu(x + s_hgt * t * rsqrt(mean t^2 + eps))
__global__ __launch_bounds__(128) void k_hgt_mix(const float* __restrict__ x,
                                                 const float* __restrict__ y,
                                                 const float* __restrict__ skip,
                                                 const float* __restrict__ scale,
                                                 float* __restrict__ xout, int N) {
  const int n = blockIdx.x, o = threadIdx.x;
  __shared__ float red[128];
  float g = 1.f / (1.f + expf(-skip[0]));
  float xv = x[(size_t)n * HD + o];
  float t = g * y[(size_t)n * HD + o] + (1.f - g) * xv;
  red[o] = t * t;
  __syncthreads();
  for (int st = 64; st > 0; st >>= 1) { if (o < st) red[o] += red[o + st]; __syncthreads(); }
  float r = rsqrtf(red[0] / 128.f + EPS_RMS);
  xout[(size_t)n * HD + o] = lrelu_f(xv + scale[o] * t * r);
}

// ---------------- GAT edge pipeline ----------------
// a_s[n,h] = sum_j h[n,h*64+j]*att_src[h,j]; a_d analogous.
__global__ __launch_bounds__(128) void k_gat_att(const float* __restrict__ h,
                                                 const float* __restrict__ att_src,
                                                 const float* __restrict__ att_dst,
                                                 float* __restrict__ a_s,
                                                 float* __restrict__ a_d, int N) {
  const int n = blockIdx.x, t = threadIdx.x;
  const int hd = t >> 6, j = t & 63;
  __shared__ float rs[128], rd[128];
  float v = h[(size_t)n * HD + t];
  rs[t] = v * att_src[t];
  rd[t] = v * att_dst[t];
  __syncthreads();
  for (int st = 32; st > 0; st >>= 1) {
    if (j < st) { rs[t] += rs[t + st]; rd[t] += rd[t + st]; }
    __syncthreads();
  }
  if (j == 0) { a_s[n * GAT_H + hd] = rs[t]; a_d[n * GAT_H + hd] = rd[t]; }
}

__global__ void k_gat_logit(const int* __restrict__ src, const int* __restrict__ dst,
                            int E, int ET, const float* __restrict__ a_s,
                            const float* __restrict__ a_d, float* __restrict__ logits,
                            unsigned* __restrict__ mEnc) {
  int e = blockIdx.x * blockDim.x + threadIdx.x;
  if (e >= ET) return;
  int sN, dN;
  if (e < E) { sN = src[e]; dN = dst[e]; } else { sN = dN = e - E; }
#pragma unroll
  for (int hh = 0; hh < GAT_H; ++hh) {
    float l = lrelu_f(a_s[sN * GAT_H + hh] + a_d[dN * GAT_H + hh]);
    logits[(size_t)e * GAT_H + hh] = l;
    atomicMax(&mEnc[dN * GAT_H + hh], enc_f(l));
  }
}

__global__ void k_gat_expsum(const int* __restrict__ dst, int E, int ET,
                             const unsigned* __restrict__ mEnc,
                             float* __restrict__ logits, float* __restrict__ ssum) {
  int e = blockIdx.x * blockDim.x + threadIdx.x;
  if (e >= ET) return;
  int dN = (e < E) ? dst[e] : (e - E);
#pragma unroll
  for (int hh = 0; hh < GAT_H; ++hh) {
    float m = dec_f(mEnc[dN * GAT_H + hh]);
    float ex = expf(logits[(size_t)e * GAT_H + hh] - m);
    logits[(size_t)e * GAT_H + hh] = ex;
    atomicAdd(&ssum[dN * GAT_H + hh], ex);
  }
}

// One thread = one edge x 4 columns: float4 gather (coalesced 512B/edge across 32
// lanes) + 4 atomic f32 adds. Head index constant within a float4 (4 | 64).
__global__ void k_gat_scatter(const int* __restrict__ src, const int* __restrict__ dst,
                              int E, int ET, const float* __restrict__ hbuf,
                              const float* __restrict__ logits,
                              const float* __restrict__ ssum, float* __restrict__ acc) {
  size_t idx = (size_t)blockIdx.x * blockDim.x + threadIdx.x;
  size_t tot = (size_t)ET * 32;
  if (idx >= tot) return;
  int c0 = ((int)(idx & 31)) * 4;
  int e = (int)(idx >> 5);
  int sN, dN;
  if (e < E) { sN = src[e]; dN = dst[e]; } else { sN = dN = e - E; }
  int hh = c0 >> 6;
  float w = logits[(size_t)e * GAT_H + hh] / (ssum[dN * GAT_H + hh] + 1e-16f);
  const float4 hv = *(const float4*)(hbuf + (size_t)sN * HD + c0);
  float* ap = acc + (size_t)dN * HD + c0;
  atomicAdd(ap + 0, hv.x * w);
  atomicAdd(ap + 1, hv.y * w);
  atomicAdd(ap + 2, hv.z * w);
  atomicAdd(ap + 3, hv.w * w);
}

// ---------------- HGT edge pipeline ----------------
__global__ __launch_bounds__(128) void k_hgt_logit(const int* __restrict__ src,
                                                   const int* __restrict__ dst, int E,
                                                   const float* __restrict__ q,
                                                   const float* __restrict__ kr,
                                                   const float* __restrict__ p_rel,
                                                   float* __restrict__ logits,
                                                   unsigned* __restrict__ mEnc) {
  const int e = blockIdx.x;
  const int t = threadIdx.x;
  const int hh = t >> 5, d = t & 31;
  const int sN = src[e], dN = dst[e];
  float pr = q[(size_t)dN * HD + t] * kr[(size_t)sN * HD + t];
  for (int off = 16; off > 0; off >>= 1) pr += __shfl_down(pr, off, 32);
  if (d == 0) {
    float l = pr * p_rel[hh] * INV_SQRT32;
    logits[(size_t)e * HGT_H + hh] = l;
    atomicMax(&mEnc[dN * HGT_H + hh], enc_f(l));
  }
}

__global__ void k_hgt_expsum(const int* __restrict__ dst, int E,
                             const unsigned* __restrict__ mEnc,
                             float* __restrict__ logits, float* __restrict__ ssum) {
  int idx = blockIdx.x * blockDim.x + threadIdx.x;
  if (idx >= E * HGT_H) return;
  int e = idx >> 2, hh = idx & 3;
  int dN = dst[e];
  float m = dec_f(mEnc[dN * HGT_H + hh]);
  float ex = expf(logits[(size_t)e * HGT_H + hh] - m);
  logits[(size_t)e * HGT_H + hh] = ex;
  atomicAdd(&ssum[dN * HGT_H + hh], ex);
}

__global__ void k_hgt_scatter(const int* __restrict__ src, const int* __restrict__ dst,
                              int E, const float* __restrict__ vr,
                              const float* __restrict__ logits,
                              const float* __restrict__ ssum, float* __restrict__ acc) {
  size_t idx = (size_t)blockIdx.x * blockDim.x + threadIdx.x;
  size_t tot = (size_t)E * 32;
  if (idx >= tot) return;
  int c0 = ((int)(idx & 31)) * 4;
  int e = (int)(idx >> 5);
  int sN = src[e], dN = dst[e];
  int hh = c0 >> 5;
  float w = logits[(size_t)e * HGT_H + hh] / (ssum[dN * HGT_H + hh] + 1e-16f);
  const float4 vv = *(const float4*)(vr + (size_t)sN * HD + c0);
  float* ap = acc + (size_t)dN * HD + c0;
  atomicAdd(ap + 0, vv.x * w);
  atomicAdd(ap + 1, vv.y * w);
  atomicAdd(ap + 2, vv.z * w);
  atomicAdd(ap + 3, vv.w * w);
}

// ---------------- FiLM ----------------
__global__ __launch_bounds__(256) void k_film_gb(const float* __restrict__ z,
                                                 const float* __restrict__ W1,
                                                 const float* __restrict__ b1,
                                                 const float* __restrict__ W2,
                                                 const float* __restrict__ b2,
                                                 float* __restrict__ fg,
                                                 float* __restrict__ fb) {
  const int b = blockIdx.x, o = threadIdx.x;  // o in [0,256)
  __shared__ float zs[128];
  __shared__ float h1[256];
  if (o < 128) zs[o] = z[b * HD + o];
  __syncthreads();
  float s = b1[o];
  for (int i = 0; i < 128; ++i) s += zs[i] * W1[o * 128 + i];
  h1[o] = gelu_f(s);
  __syncthreads();
  float s2 = b2[o];
  for (int i = 0; i < 256; ++i) s2 += h1[i] * W2[o * 256 + i];
  float v = 0.1f * tanhf(s2);
  if (o < 128) fg[b * HD + o] = v; else fb[b * HD + (o - 128)] = v;
}

__global__ void k_film_apply(float* __restrict__ x, const float* __restrict__ fg,
                             const float* __restrict__ fb, int C, size_t tot4) {
  size_t idx = (size_t)blockIdx.x * blockDim.x + threadIdx.x;
  if (idx >= tot4) return;
  int c4 = (int)(idx & 31);
  int n = (int)(idx >> 5);
  int b = n / C;
  float4 xv = ((float4*)x)[idx];
  float4 g = ((const float4*)fg)[b * 32 + c4];
  float4 bb = ((const float4*)fb)[b * 32 + c4];
  xv.x = (1.f + g.x) * xv.x + bb.x;
  xv.y = (1.f + g.y) * xv.y + bb.y;
  xv.z = (1.f + g.z) * xv.z + bb.z;
  xv.w = (1.f + g.w) * xv.w + bb.w;
  ((float4*)x)[idx] = xv;
}

// ---------------- hmod ----------------
__global__ __launch_bounds__(128) void k_hmod_q(const float* __restrict__ z,
                                                const float* __restrict__ Wq,
                                                const float* __restrict__ bq,
                                                float* __restrict__ q) {
  const int b = blockIdx.x, o = threadIdx.x;
  __shared__ float zs[128];
  zs[o] = z[b * HD + o];
  __syncthreads();
  float s = bq[o];
  for (int i = 0; i < 128; ++i) s += zs[i] * Wq[o * 128 + i];
  q[b * HD + o] = s;
}

__global__ __launch_bounds__(256) void k_hmod_attn(const float* __restrict__ q,
                                                   const float* __restrict__ k,
                                                   const float* __restrict__ v,
                                                   float* __restrict__ ctx, int C) {
  if (C > 1024) return;
  const int b = blockIdx.x >> 2;
  const int hh = blockIdx.x & 3;
  const int t = threadIdx.x;
  __shared__ float sc[1024];
  __shared__ float red[256];
  __shared__ float qs[32];
  if (t < 32) qs[t] = q[b * HD + hh * 32 + t];
  __syncthreads();
  float lmax = -3.4e38f;
  for (int c = t; c < C; c += 256) {
    const float* kp = k + (size_t)(b * C + c) * HD + hh * 32;
    float s = 0.f;
    for (int d = 0; d < 32; ++d) s += qs[d] * kp[d];
    s *= INV_SQRT32;
    sc[c] = s;
    lmax = fmaxf(lmax, s);
  }
  red[t] = lmax;
  __syncthreads();
  for (int st = 128; st > 0; st >>= 1) { if (t < st) red[t] = fmaxf(red[t], red[t + st]); __syncthreads(); }
  float mx = red[0];
  __syncthreads();
  float lsum = 0.f;
  for (int c = t; c < C; c += 256) { float e = expf(sc[c] - mx); sc[c] = e; lsum += e; }
  red[t] = lsum;
  __syncthreads();
  for (int st = 128; st > 0; st >>= 1) { if (t < st) red[t] += red[t + st]; __syncthreads(); }
  float tot = red[0];
  __syncthreads();
  const int d = t & 31, part = t >> 5;
  float a0 = 0.f;
  for (int c = part; c < C; c += 8)
    a0 += sc[c] * v[(size_t)(b * C + c) * HD + hh * 32 + d];
  red[t] = a0;
  __syncthreads();
  if (part == 0) {
    float s = 0.f;
    for (int pp = 0; pp < 8; ++pp) s += red[pp * 32 + d];
    ctx[b * HD + hh * 32 + d] = s / tot;
  }
}

__global__ __launch_bounds__(128) void k_hmod_mlp(const float* __restrict__ ctx,
                                                  float* __restrict__ z,
                                                  const float* __restrict__ Wm1,
                                                  const float* __restrict__ bm1,
                                                  const float* __restrict__ Wm2,
                                                  const float* __restrict__ bm2,
                                                  const float* __restrict__ scale) {
  const int b = blockIdx.x, o = threadIdx.x;
  __shared__ float zin[256];
  __shared__ float h1[128];
  __shared__ float red[128];
  zin[o] = ctx[b * HD + o];
  zin[128 + o] = z[b * HD + o];
  __syncthreads();
  float s = bm1[o];
  for (int i = 0; i < 256; ++i) s += zin[i] * Wm1[o * 256 + i];
  h1[o] = gelu_f(s);
  __syncthreads();
  float s2 = bm2[o];
  for (int i = 0; i < 128; ++i) s2 += h1[i] * Wm2[o * 128 + i];
  float t = zin[128 + o] + s2;
  red[o] = t * t;
  __syncthreads();
  for (int st = 64; st > 0; st >>= 1) { if (o < st) red[o] += red[o + st]; __syncthreads(); }
  float r = rsqrtf(red[0] / 128.f + EPS_RMS);
  z[b * HD + o] = scale[o] * t * r;
}

// ---------------- readout ----------------
__global__ void k_cat(const float* __restrict__ x, const float* __restrict__ z,
                      float* __restrict__ cat, int C, size_t tot4) {
  size_t idx = (size_t)blockIdx.x * blockDim.x + threadIdx.x;
  if (idx >= tot4) return;
  int j4 = (int)(idx & 63);
  int n = (int)(idx >> 6);
  float4 v;
  if (j4 < 32) v = ((const float4*)x)[(size_t)n * 32 + j4];
  else v = ((const float4*)z)[(n / C) * 32 + (j4 - 32)];
  ((float4*)cat)[idx] = v;
}

__global__ __launch_bounds__(128) void k_readout(const float* __restrict__ h,
                                                 const float* __restrict__ W2,
                                                 const float* __restrict__ b2,
                                                 float* __restrict__ out, int N) {
  const int n = blockIdx.x, o = threadIdx.x;
  __shared__ float red[128];
  red[o] = h[(size_t)n * HD + o] * W2[o];
  __syncthreads();
  for (int st = 64; st > 0; st >>= 1) { if (o < st) red[o] += red[o + st]; __syncthreads(); }
  if (o == 0) out[n] = red[0] + b2[0];
}

// ==================================================================================
extern "C" void kernel_launch(void* const* d_in, const int* in_sizes, int n_in,
                              void* d_out, int out_size, void* d_ws, size_t ws_size,
                              hipStream_t stream) {
  if (n_in < 69) return;
  const int N = in_sizes[0] / 16;
  const int E = in_sizes[1] / 2;
  const int B = 50;                 // BATCH module constant
  const int C = N / B;
  const int ET = E + N;             // edges + self loops (GAT)
  const size_t S = (size_t)N * HD;

#define PL(i) ((const float*)d_in[5 + (i)])
  const float* xc = (const float*)d_in[0];
  const int* eiS[3] = {(const int*)d_in[1], (const int*)d_in[2], (const int*)d_in[3]};

  // workspace layout
  size_t needF = 9 * S + 12 * (size_t)N + 4 * (size_t)ET + 640 * (size_t)B;
  if (ws_size < needF * sizeof(float)) return;
  float* W0 = (float*)d_ws;
  float* xemb = W0;
  float* x = W0 + S;
  float* bQ = W0 + 2 * S;
  float* bK = W0 + 3 * S;
  float* bV = W0 + 4 * S;
  float* bKR = W0 + 5 * S;   // also GAT h buffer; with bVR forms the [N,256] cat buffer
  float* bVR = W0 + 6 * S;
  float* bACC = W0 + 7 * S;
  float* bTMP = W0 + 8 * S;
  float* statM = W0 + 9 * S;                 // N*4 (unsigned encodings)
  float* statS = statM + 4 * (size_t)N;      // N*4
  float* asad = statS + 4 * (size_t)N;       // a_s [N*2] then a_d [N*2]
  float* logit = asad + 4 * (size_t)N;       // ET*4
  float* zbuf = logit + 4 * (size_t)ET;      // B*128
  float* ctx = zbuf + B * HD;
  float* fg = ctx + B * HD;
  float* fb = fg + B * HD;
  float* qh = fb + B * HD;

  const int gGemm = (N + 127) / 128;
  const int gBD = (N + 15) / 16;
  const int gS = (int)((S + 255) / 256);
  const int gS4 = (int)((S / 4 + 255) / 256);

  auto gat_layer = [&](int base, const float* scale) {
    k_bias3<<<gS, 256, 0, stream>>>(PL(base + 3), PL(base + 7), PL(base + 11), bACC, S);
    for (int r = 0; r < 3; ++r) {
      const int* src = eiS[r];
      const int* dst = eiS[r] + E;
      k_gemm<128, 0><<<gGemm, 256, 0, stream>>>(x, PL(base + 4 * r), nullptr, bKR, N);
      k_gat_att<<<N, 128, 0, stream>>>(bKR, PL(base + 4 * r + 2), PL(base + 4 * r + 1),
                                       asad, asad + 2 * (size_t)N, N);
      k_fill_u32<<<(N * 2 + 255) / 256, 256, 0, stream>>>((unsigned*)statM, NEG_INF_ENC, N * 2);
      k_fill_f32<<<(N * 2 + 255) / 256, 256, 0, stream>>>(statS, 0.f, (size_t)N * 2);
      k_gat_logit<<<(ET + 255) / 256, 256, 0, stream>>>(src, dst, E, ET, asad,
                                                        asad + 2 * (size_t)N, logit,
                                                        (unsigned*)statM);
      k_gat_expsum<<<(ET + 255) / 256, 256, 0, stream>>>(dst, E, ET, (unsigned*)statM,
                                                         logit, statS);
      k_gat_scatter<<<(int)(((size_t)ET * 32 + 255) / 256), 256, 0, stream>>>(
          src, dst, E, ET, bKR, logit, statS, bACC);
    }
    k_rms_res_lrelu<<<N, 128, 0, stream>>>(x, bACC, scale, x, N);
  };

  // ---- embed ----
  k_embed<<<N, 128, 0, stream>>>(xc, PL(0), PL(1), PL(2), xemb, N);
  hipMemcpyAsync(x, xemb, S * sizeof(float), hipMemcpyDeviceToDevice, stream);
  k_zinit<<<(B * HD + 255) / 256, 256, 0, stream>>>(zbuf, PL(63), B * HD);

  for (int cyc = 0; cyc < 2; ++cyc) {
    // ================= lblock =================
    gat_layer(20, PL(56));  // gat1, s1
    gat_layer(32, PL(57));  // gat2, s2

    // HGT
    k_gemm<128, MBIAS><<<gGemm, 256, 0, stream>>>(x, PL(45), PL(50), bK, N);  // Wk,bk
    k_gemm<128, MBIAS><<<gGemm, 256, 0, stream>>>(x, PL(46), PL(51), bQ, N);  // Wq,bq
    k_gemm<128, MBIAS><<<gGemm, 256, 0, stream>>>(x, PL(47), PL(52), bV, N);  // Wv,bv
    k_fill_f32<<<gS, 256, 0, stream>>>(bACC, 0.f, S);
    for (int r = 0; r < 3; ++r) {
      const int* src = eiS[r];
      const int* dst = eiS[r] + E;
      k_gemm_bd<<<gBD, 256, 0, stream>>>(bK, PL(48) + (size_t)r * 4096, bKR, N);
      k_gemm_bd<<<gBD, 256, 0, stream>>>(bV, PL(53) + (size_t)r * 4096, bVR, N);
      k_fill_u32<<<(N * 4 + 255) / 256, 256, 0, stream>>>((unsigned*)statM, NEG_INF_ENC, N * 4);
      k_fill_f32<<<(N * 4 + 255) / 256, 256, 0, stream>>>(statS, 0.f, (size_t)N * 4);
      k_hgt_logit<<<E, 128, 0, stream>>>(src, dst, E, bQ, bKR, PL(54) + r * 4, logit,
                                         (unsigned*)statM);
      k_hgt_expsum<<<(E * 4 + 255) / 256, 256, 0, stream>>>(dst, E, (unsigned*)statM,
                                                            logit, statS);
      k_hgt_scatter<<<(int)(((size_t)E * 32 + 255) / 256), 256, 0, stream>>>(
          src, dst, E, bVR, logit, statS, bACC);
    }
    k_gemm<128, MBIAS | MGELUA><<<gGemm, 256, 0, stream>>>(bACC, PL(44), PL(49), bTMP, N);
    k_hgt_mix<<<N, 128, 0, stream>>>(x, bTMP, PL(55), PL(58), x, N);
    // input injection: x += xemb @ Wi^T + bi
    k_gemm<128, MBIAS | MACC><<<gGemm, 256, 0, stream>>>(xemb, PL(14), PL(15), x, N);
    // FiLM
    k_film_gb<<<B, 256, 0, stream>>>(zbuf, PL(16), PL(18), PL(17), PL(19), fg, fb);
    k_film_apply<<<gS4, 256, 0, stream>>>(x, fg, fb, C, S / 4);

    // ================= hmod =================
    k_hmod_q<<<B, 128, 0, stream>>>(zbuf, PL(6), PL(11), qh);
    k_gemm<128, MBIAS><<<gGemm, 256, 0, stream>>>(x, PL(3), PL(8), bK, N);   // hmod Wk
    k_gemm<128, MBIAS><<<gGemm, 256, 0, stream>>>(x, PL(7), PL(12), bV, N);  // hmod Wv
    k_hmod_attn<<<B * 4, 256, 0, stream>>>(qh, bK, bV, ctx, C);
    k_hmod_mlp<<<B, 128, 0, stream>>>(ctx, zbuf, PL(4), PL(9), PL(5), PL(10), PL(13));
  }

  // ---- readout ----
  k_cat<<<(int)(((size_t)N * 64 + 255) / 256), 256, 0, stream>>>(x, zbuf, bKR, C,
                                                                 (size_t)N * 64);
  k_gemm<256, MBIAS | MGELUO><<<gGemm, 256, 0, stream>>>(bKR, PL(59), PL(61), bTMP, N);
  k_readout<<<N, 128, 0, stream>>>(bTMP, PL(60), PL(62), (float*)d_out, N);
  (void)out_size;
}